// ACTLoss_56624848831010
// MI455X (gfx1250) — compile-verified
//
#include <hip/hip_runtime.h>

// ---------------------------------------------------------------------------
// ACT loss for MI455X (gfx1250, wave32).
//
// Key observation: losses_per_step[k,b] = ce[b] + k*0.01 is strictly
// increasing in k, and argmin picks the first minimum => optimal_k == 0 for
// every sample, exactly. Therefore:
//   * logits / labels never influence the output (ce is dead) -> skip 131 MB.
//   * update_critic=1 branch has an all-false mask -> loss = 0 exactly.
//   * update_critic=0 branch: mask = (kp>0),
//       per = -0.1*kp * log( cumsum(contrib)[kp-1,b]/kp + 1e-8 )
//       loss = sum(per*mask)/max(sum(mask),1)  (0 if no masked samples).
//
// cumsum over K=16 is computed as a triangular matmul L(16x16) @ contrib
// using 4 chained V_WMMA_F32_16X16X4_F32 per 16-sample tile (one wave/tile).
// Live traffic: 2 MB -> ~90 ns at 23.3 TB/s; this is launch-bound, so one
// small WMMA kernel + one tiny deterministic reduction kernel.
// ---------------------------------------------------------------------------

typedef __attribute__((ext_vector_type(2))) float v2f;
typedef __attribute__((ext_vector_type(8))) float v8f;

__global__ void __launch_bounds__(256) act_partials_kernel(
    const float* __restrict__ contrib,   // [16, B] row-major
    const int*   __restrict__ halt,      // [B] (harness passes ints as i32)
    float* __restrict__ part_sum,        // [nwaves]
    float* __restrict__ part_cnt,        // [nwaves]
    int B)
{
    const int lane = threadIdx.x & 31;
    const int wave = (blockIdx.x * blockDim.x + threadIdx.x) >> 5;
    const int m    = lane & 15;   // row (lanes 0-15 / 16-31 both map M=0..15)
    const int h    = lane >> 4;   // lane half selects K pairs
    const int n    = wave * 16 + m;           // sample / column index

    __builtin_prefetch(&halt[n], 0, 3);       // -> global_prefetch_b8

    // cumc(16x16 tile) = L @ contrib_tile via 4x V_WMMA_F32_16X16X4_F32.
    // A layout (16x4 f32): lane l, VGPR j  -> A[l&15][2*(l>>4)+j]
    // B layout (4x16 f32): lane l, VGPR j  -> B[2*(l>>4)+j][l&15]
    // D layout (16x16 f32): lane l, VGPR r -> D[r + 8*(l>>4)][l&15]
    v8f acc = {};
#pragma unroll
    for (int kk = 0; kk < 4; ++kk) {
        const int k0 = kk * 4 + h * 2;        // global K index of this lane's pair
        v2f a, b;
        a.x = (k0     <= m) ? 1.0f : 0.0f;    // lower-triangular ones (incl diag)
        a.y = (k0 + 1 <= m) ? 1.0f : 0.0f;
        b.x = contrib[(size_t)(k0    ) * B + n];
        b.y = contrib[(size_t)(k0 + 1) * B + n];
        acc = __builtin_amdgcn_wmma_f32_16x16x4_f32(
            /*neg_a=*/false, a, /*neg_b=*/false, b,
            /*c_mod=*/(short)0, acc, /*reuse_a=*/false, /*reuse_b=*/false);
    }
    // acc[r] == cumc[r + 8*h][n]

    const int kp  = halt[n];                  // 0..16
    int idx = kp - 1;
    if (idx < 0)  idx = 0;
    if (idx > 15) idx = 15;

    // Row idx lives in lane-half (idx>>3); combine halves so all lanes get it.
    float cand = (h == (idx >> 3)) ? acc[idx & 7] : 0.0f;
    float csel = cand + __shfl_xor(cand, 16, 32);   // cumc[idx][n]

    float per = 0.0f, msk = 0.0f;             // per already includes mask (0 if kp<=0)
    if (kp > 0) {
        msk = 1.0f;
        float mean_c = csel / (float)kp;      // maximum(kp,1) == kp here
        // advantage = (0 - kp)*(-0.1) = 0.1*kp ;  per = -advantage*log(mean_c+1e-8)
        per = -0.1f * (float)kp * logf(mean_c + 1e-8f);
    }

    // Count each sample once (half h==0), then deterministic wave reduce.
    float s = (h == 0) ? per : 0.0f;
    float c = (h == 0) ? msk : 0.0f;
#pragma unroll
    for (int off = 16; off >= 1; off >>= 1) {
        s += __shfl_xor(s, off, 32);
        c += __shfl_xor(c, off, 32);
    }
    if (lane == 0) {
        part_sum[wave] = s;
        part_cnt[wave] = c;
    }
}

__global__ void __launch_bounds__(256) act_finalize_kernel(
    const float* __restrict__ part_sum,
    const float* __restrict__ part_cnt,
    const int*   __restrict__ update_critic,
    float* __restrict__ out,
    int nwaves)
{
    __shared__ float ss[256];
    __shared__ float sc[256];
    const int t = threadIdx.x;
    float s = 0.0f, c = 0.0f;
    for (int i = t; i < nwaves; i += 256) {   // fixed order -> deterministic
        s += part_sum[i];
        c += part_cnt[i];
    }
    ss[t] = s; sc[t] = c;
    __syncthreads();
#pragma unroll
    for (int off = 128; off >= 1; off >>= 1) {
        if (t < off) { ss[t] += ss[t + off]; sc[t] += sc[t + off]; }
        __syncthreads();
    }
    if (t == 0) {
        float loss = (sc[0] > 0.0f) ? (ss[0] / fmaxf(sc[0], 1.0f)) : 0.0f;
        // Critic branch: optimal_k==0 => mask all-false => loss is exactly 0.
        if (update_critic[0] != 0) loss = 0.0f;
        out[0] = loss;
    }
}

extern "C" void kernel_launch(void* const* d_in, const int* in_sizes, int n_in,
                              void* d_out, int out_size, void* d_ws, size_t ws_size,
                              hipStream_t stream)
{
    // Inputs: 0=logits(dead) 1=labels(dead) 2=contributions[16,B]
    //         3=thresholds(dead) 4=halt_iterations[B] 5=update_critic[1]
    const float* contrib = (const float*)d_in[2];
    const int*   halt    = (const int*)d_in[4];
    const int*   uc      = (const int*)d_in[5];
    float*       out     = (float*)d_out;

    const int K      = 16;
    const int B      = in_sizes[2] / K;       // 32768
    const int nwaves = B / 16;                // one wave per 16 samples
    const int block  = 256;                   // 8 waves/block (wave32)
    const int blocks = nwaves / (block / 32);

    float* part_sum = (float*)d_ws;           // nwaves floats
    float* part_cnt = part_sum + nwaves;      // nwaves floats (16 KB total)

    act_partials_kernel<<<blocks, block, 0, stream>>>(contrib, halt,
                                                      part_sum, part_cnt, B);
    act_finalize_kernel<<<1, 256, 0, stream>>>(part_sum, part_cnt, uc,
                                               out, nwaves);
}